// mwt_transform_cross_3487513444897
// MI455X (gfx1250) — compile-verified
//
#include <hip/hip_runtime.h>
#include <hip/hip_bf16.h>

typedef __attribute__((ext_vector_type(2))) float v2f;
typedef __attribute__((ext_vector_type(8))) float v8f;

#define WPB 4 // waves (32x32 macrotiles) per 128-thread block

// ---------------------------------------------------------------------------
// Generic strided, batched f32 GEMM on the CDNA5 matrix pipe.
//   C[p] = alpha * A[p] x B[p] + beta * C[p] (+ bias[col])
// Element addressing: A(r,k) = A + r*ars + k*acs  (plus 2-level batch strides)
// V_WMMA_F32_16X16X4_F32; one wave owns a 32x32 macrotile (2x2 register tile,
// 4 accumulators). Per K-step: 8 loads -> 4 WMMAs (2 loads/WMMA), independent
// accumulator chains, software-pipelined (next operands loaded before current
// WMMAs). OOB rows/cols are CLAMPED (their products land in never-stored
// lanes); only the single K-tail step is masked, branchlessly.
// ---------------------------------------------------------------------------
__global__ void __launch_bounds__(128)
gemm_wmma_f32(const float* __restrict__ A, const float* __restrict__ Bm,
              float* __restrict__ Cm, const float* __restrict__ bias,
              int Mdim, int Ndim, int Kdim,
              long ars, long acs, long brs, long bcs, long crs, long ccs,
              long ab1, long ab2, long bb1, long bb2, long cb1, long cb2,
              int nb2, float alpha, float beta)
{
  const int lane   = threadIdx.x & 31;
  const int wave   = threadIdx.x >> 5;
  const int tilesN = (Ndim + 31) >> 5;
  const int tilesM = (Mdim + 31) >> 5;
  const int tile   = blockIdx.x * WPB + wave;
  if (tile >= tilesM * tilesN) return;
  const int tm = tile / tilesN;
  const int tn = tile - tm * tilesN;
  const int p  = blockIdx.y;
  const int b1 = p / nb2;
  const int b2 = p - b1 * nb2;
  const float* Ab = A  + (long)b1 * ab1 + (long)b2 * ab2;
  const float* Bb = Bm + (long)b1 * bb1 + (long)b2 * bb2;
  float*       Cb = Cm + (long)b1 * cb1 + (long)b2 * cb2;

  // A lane layout (16x4 f32): lanes 0-15 / 16-31 both hold rows 0-15,
  // lane>>4 selects k-pair {0,1} vs {2,3}. Same mapping for B columns.
  const int l15 = lane & 15;
  const int kp  = (lane >> 4) << 1;
  const int rA0 = tm * 32 + l15;
  const int rA1 = rA0 + 16;
  const int cB0 = tn * 32 + l15;
  const int cB1 = cB0 + 16;
  // Clamp OOB rows/cols into range (results land in never-stored lanes).
  const int rA0c = (rA0 < Mdim) ? rA0 : (Mdim - 1);
  const int rA1c = (rA1 < Mdim) ? rA1 : (Mdim - 1);
  const int cB0c = (cB0 < Ndim) ? cB0 : (Ndim - 1);
  const int cB1c = (cB1 < Ndim) ? cB1 : (Ndim - 1);

  const float* pA0 = Ab + (long)rA0c * ars + (long)kp * acs;
  const float* pA1 = Ab + (long)rA1c * ars + (long)kp * acs;
  const float* pB0 = Bb + (long)cB0c * bcs + (long)kp * brs;
  const float* pB1 = Bb + (long)cB1c * bcs + (long)kp * brs;
  const long aStep = 4 * acs;
  const long bStep = 4 * brs;

  v8f acc00 = {0.f, 0.f, 0.f, 0.f, 0.f, 0.f, 0.f, 0.f};
  v8f acc01 = acc00, acc10 = acc00, acc11 = acc00;

  const int kmain = Kdim & ~3;
  if (kmain > 0) {
    v2f a0, a1, b0, b1;
    a0.x = pA0[0]; a0.y = pA0[acs];
    a1.x = pA1[0]; a1.y = pA1[acs];
    b0.x = pB0[0]; b0.y = pB0[brs];
    b1.x = pB1[0]; b1.y = pB1[brs];
    for (int k0 = 4; k0 < kmain; k0 += 4) {
      pA0 += aStep; pA1 += aStep; pB0 += bStep; pB1 += bStep;
      v2f na0, na1, nb0, nb1;
      na0.x = pA0[0]; na0.y = pA0[acs];
      na1.x = pA1[0]; na1.y = pA1[acs];
      nb0.x = pB0[0]; nb0.y = pB0[brs];
      nb1.x = pB1[0]; nb1.y = pB1[brs];
      acc00 = __builtin_amdgcn_wmma_f32_16x16x4_f32(false, a0, false, b0,
                                                    (short)0, acc00, false, false);
      acc01 = __builtin_amdgcn_wmma_f32_16x16x4_f32(false, a0, false, b1,
                                                    (short)0, acc01, false, false);
      acc10 = __builtin_amdgcn_wmma_f32_16x16x4_f32(false, a1, false, b0,
                                                    (short)0, acc10, false, false);
      acc11 = __builtin_amdgcn_wmma_f32_16x16x4_f32(false, a1, false, b1,
                                                    (short)0, acc11, false, false);
      a0 = na0; a1 = na1; b0 = nb0; b1 = nb1;
    }
    acc00 = __builtin_amdgcn_wmma_f32_16x16x4_f32(false, a0, false, b0,
                                                  (short)0, acc00, false, false);
    acc01 = __builtin_amdgcn_wmma_f32_16x16x4_f32(false, a0, false, b1,
                                                  (short)0, acc01, false, false);
    acc10 = __builtin_amdgcn_wmma_f32_16x16x4_f32(false, a1, false, b0,
                                                  (short)0, acc10, false, false);
    acc11 = __builtin_amdgcn_wmma_f32_16x16x4_f32(false, a1, false, b1,
                                                  (short)0, acc11, false, false);
  }
  if (kmain < Kdim) { // single masked K-tail step (branchless)
    const int   ka  = kmain + kp;
    const int   kb  = ka + 1;
    const int   kac = (ka < Kdim) ? ka : (Kdim - 1);
    const int   kbc = (kb < Kdim) ? kb : (Kdim - 1);
    const float mka = (ka < Kdim) ? 1.f : 0.f;
    const float mkb = (kb < Kdim) ? 1.f : 0.f;
    v2f a0, a1, b0, b1;
    a0.x = Ab[(long)rA0c * ars + (long)kac * acs] * mka; // masking A alone
    a0.y = Ab[(long)rA0c * ars + (long)kbc * acs] * mkb; // zeroes the product
    a1.x = Ab[(long)rA1c * ars + (long)kac * acs] * mka;
    a1.y = Ab[(long)rA1c * ars + (long)kbc * acs] * mkb;
    b0.x = Bb[(long)cB0c * bcs + (long)kac * brs];
    b0.y = Bb[(long)cB0c * bcs + (long)kbc * brs];
    b1.x = Bb[(long)cB1c * bcs + (long)kac * brs];
    b1.y = Bb[(long)cB1c * bcs + (long)kbc * brs];
    acc00 = __builtin_amdgcn_wmma_f32_16x16x4_f32(false, a0, false, b0,
                                                  (short)0, acc00, false, false);
    acc01 = __builtin_amdgcn_wmma_f32_16x16x4_f32(false, a0, false, b1,
                                                  (short)0, acc01, false, false);
    acc10 = __builtin_amdgcn_wmma_f32_16x16x4_f32(false, a1, false, b0,
                                                  (short)0, acc10, false, false);
    acc11 = __builtin_amdgcn_wmma_f32_16x16x4_f32(false, a1, false, b1,
                                                  (short)0, acc11, false, false);
  }

  // C layout: VGPR r, lanes 0-15 -> (M=r, N=l15), lanes 16-31 -> (M=r+8).
  const int rb = (lane >> 4) << 3;
  auto store_tile = [&](const v8f& acc, int row0, int col) {
    if (col >= Ndim) return;
    const float bval = bias ? bias[col] : 0.f;
#pragma unroll
    for (int r = 0; r < 8; ++r) {
      const int row = row0 + r;
      if (row < Mdim) {
        const long off = (long)row * crs + (long)col * ccs;
        float v = alpha * acc[r] + bval;
        if (beta != 0.f) v += beta * Cb[off];
        Cb[off] = v;
      }
    }
  };
  store_tile(acc00, tm * 32 + rb,      cB0);
  store_tile(acc01, tm * 32 + rb,      cB1);
  store_tile(acc10, tm * 32 + 16 + rb, cB0);
  store_tile(acc11, tm * 32 + 16 + rb, cB1);
}

// ---------------------------------------------------------------------------
// Wavelet decomposition step: x (B, 2*Lh, 64, 8) -> d,s (B, Lh, 64, 8)
// d[k] = sum_j x_even[j]*ec_d[j,k] + x_odd[j]*ec_d[8+j,k]   (same for s)
// ---------------------------------------------------------------------------
__global__ void wavelet_step(const float* __restrict__ x,
                             float* __restrict__ dO, float* __restrict__ sO,
                             const float* __restrict__ ecd,
                             const float* __restrict__ ecs, int Bn, int Lh)
{
  long idx = (long)blockIdx.x * blockDim.x + threadIdx.x;
  long tot = (long)Bn * Lh * 64;
  if (idx >= tot) return;
  int  c  = (int)(idx % 64);
  long t  = idx / 64;
  int  np = (int)(t % Lh);
  long b  = t / Lh;
  const float* xe = x + (((long)b * (2 * Lh) + 2 * np) * 64 + c) * 8;
  const float* xo = xe + 512; // next n row
  float xl[16];
#pragma unroll
  for (int j = 0; j < 8; ++j) { xl[j] = xe[j]; xl[8 + j] = xo[j]; }
  long ob = (((long)b * Lh + np) * 64 + c) * 8;
#pragma unroll
  for (int kq = 0; kq < 8; ++kq) {
    float ad = 0.f, as = 0.f;
#pragma unroll
    for (int j = 0; j < 16; ++j) {
      ad += xl[j] * ecd[j * 8 + kq];
      as += xl[j] * ecs[j * 8 + kq];
    }
    dO[ob + kq] = ad;
    sO[ob + kq] = as;
  }
}

// ---------------------------------------------------------------------------
// Fused reconstruction step: xin = [ vin + Us | Ud ] (B, L, 64, 16)
// out[2n]   = xin @ rc_e ; out[2n+1] = xin @ rc_o   -> (B, 2L, 64, 8)
// ---------------------------------------------------------------------------
__global__ void recon_step(const float* __restrict__ vin,
                           const float* __restrict__ Us,
                           const float* __restrict__ Ud,
                           float* __restrict__ vout,
                           const float* __restrict__ rce,
                           const float* __restrict__ rco, int Bn, int Ln)
{
  long idx = (long)blockIdx.x * blockDim.x + threadIdx.x;
  long tot = (long)Bn * Ln * 64;
  if (idx >= tot) return;
  int  c = (int)(idx % 64);
  long t = idx / 64;
  int  n = (int)(t % Ln);
  long b = t / Ln;
  long ib = (((long)b * Ln + n) * 64 + c) * 8;
  float xl[16];
#pragma unroll
  for (int j = 0; j < 8; ++j) {
    xl[j]     = vin[ib + j] + Us[ib + j];
    xl[8 + j] = Ud[ib + j];
  }
  long ob = (((long)b * (2 * Ln) + 2 * n) * 64 + c) * 8;
#pragma unroll
  for (int kq = 0; kq < 8; ++kq) {
    float e = 0.f, o = 0.f;
#pragma unroll
    for (int j = 0; j < 16; ++j) {
      e += xl[j] * rce[j * 8 + kq];
      o += xl[j] * rco[j * 8 + kq];
    }
    vout[ob + kq]       = e;
    vout[ob + 512 + kq] = o; // row 2n+1
  }
}

// ---------------------------------------------------------------------------
// DFT twiddle tables.
// Ff (L x 2M): [cos | -sin] of 2*pi*m*t/L           (forward rfft, no norm)
// Fi (2M x L): w_m/(L*4096) * [cos ; -sin]          (irfft + 1/(64*64) scale)
// ---------------------------------------------------------------------------
__global__ void build_twiddle(float* __restrict__ Ff, float* __restrict__ Fi,
                              int L, int M)
{
  long idx = (long)blockIdx.x * blockDim.x + threadIdx.x;
  if (idx >= (long)L * M) return;
  int m = (int)(idx % M);
  int t = (int)(idx / M);
  float th = -6.2831853071795864f * (float)m * (float)t / (float)L;
  float s, c;
  __sincosf(th, &s, &c); // c = cos(theta), s = -sin(theta)
  Ff[(long)t * 2 * M + m]     = c;
  Ff[(long)t * 2 * M + M + m] = s;
  float w  = (m == 0) ? 1.f : 2.f;
  float sc = w / ((float)L * 4096.f);
  Fi[(long)m * L + t]       = sc * c;
  Fi[((long)M + m) * L + t] = sc * s;
}

// ---------------------------------------------------------------------------
// Complex tanh on G (per batch p: M rows x 2M cols, [real | imag]).
// tanh(a+ib) = (sinh 2a + i sin 2b) / (cosh 2a + cos 2b)
// ---------------------------------------------------------------------------
__global__ void ctanh_kernel(float* __restrict__ G, int nb, int M)
{
  long idx = (long)blockIdx.x * blockDim.x + threadIdx.x;
  long tot = (long)nb * M * M;
  if (idx >= tot) return;
  int  y = (int)(idx % M);
  long t = idx / M;
  int  x = (int)(t % M);
  long p = t / M;
  float* g = G + p * (long)M * 2 * M + (long)x * 2 * M;
  float a  = g[y], b = g[M + y];
  float ta = fminf(fmaxf(2.f * a, -40.f), 40.f);
  float e  = __expf(ta), ei = 1.f / e;
  float sh = 0.5f * (e - ei), ch = 0.5f * (e + ei);
  float sb, cb;
  __sincosf(2.f * b, &sb, &cb);
  float d  = ch + cb;
  g[y]     = sh / d;
  g[M + y] = sb / d;
}

__global__ void fill_f32(float* __restrict__ p, long n, float v)
{
  long i = (long)blockIdx.x * blockDim.x + threadIdx.x;
  if (i < n) p[i] = v;
}

__global__ void copy_f32(float* __restrict__ d, const float* __restrict__ s, long n)
{
  long i = (long)blockIdx.x * blockDim.x + threadIdx.x;
  if (i < n) d[i] = s[i];
}

// ---------------------------------------------------------------------------
// Host-side helpers
// ---------------------------------------------------------------------------
static inline void gemm(hipStream_t st, const float* A, const float* B, float* C,
                        const float* bias, int M, int N, int K,
                        long ars, long acs, long brs, long bcs, long crs, long ccs,
                        long ab1, long ab2, long bb1, long bb2, long cb1, long cb2,
                        int nb1, int nb2, float alpha, float beta)
{
  int tiles = ((M + 31) / 32) * ((N + 31) / 32);
  dim3 grid((tiles + WPB - 1) / WPB, nb1 * nb2, 1);
  gemm_wmma_f32<<<grid, 32 * WPB, 0, st>>>(A, B, C, bias, M, N, K,
                                           ars, acs, brs, bcs, crs, ccs,
                                           ab1, ab2, bb1, bb2, cb1, cb2,
                                           nb2, alpha, beta);
}

// _spectral_cross(xq, xk) -> outp (beta_out selects overwrite vs accumulate).
// xq/xk/outp: blocks (32, L, 64, 8). All stages except the complex tanh are WMMA GEMMs (11 calls).
static void spectral_pass(hipStream_t st, const float* xq, const float* xk,
                          float* outp, float beta_out,
                          float* Qf, float* Kf, float* G, float* QKV,
                          const float* Ff, const float* Fi, int L, int M)
{
  const int  E = 64, H = 8, Bn = 32;
  const int  M2i = 2 * M;
  const long M2  = M2i;
  const long qfu = (long)E * M2; // per-(b,h) unit of Qf/Kf/QKV
  const long gu  = (long)M * M2; // per-(b,h) unit of G
  const long xb  = (long)L * 512;

  // Forward DFT (kept modes): Qf/Kf (E x 2M) = X(E x L) @ Ff(L x 2M)
  gemm(st, xq, Ff, Qf, nullptr, E, M2i, L, 8, 512, M2, 1, M2, 1,
       xb, 1, 0, 0, 8 * qfu, qfu, Bn, H, 1.f, 0.f);
  gemm(st, xk, Ff, Kf, nullptr, E, M2i, L, 8, 512, M2, 1, M2, 1,
       xb, 1, 0, 0, 8 * qfu, qfu, Bn, H, 1.f, 0.f);
  // G = Qf^T Kf (complex, M x M): Gr = QrKr - QiKi ; Gi = QrKi + QiKr
  gemm(st, Qf, Kf, G, nullptr, M, M, E, 1, M2, M2, 1, M2, 1,
       8 * qfu, qfu, 8 * qfu, qfu, 8 * gu, gu, Bn, H, 1.f, 0.f);
  gemm(st, Qf + M, Kf + M, G, nullptr, M, M, E, 1, M2, M2, 1, M2, 1,
       8 * qfu, qfu, 8 * qfu, qfu, 8 * gu, gu, Bn, H, -1.f, 1.f);
  gemm(st, Qf, Kf + M, G + M, nullptr, M, M, E, 1, M2, M2, 1, M2, 1,
       8 * qfu, qfu, 8 * qfu, qfu, 8 * gu, gu, Bn, H, 1.f, 0.f);
  gemm(st, Qf + M, Kf, G + M, nullptr, M, M, E, 1, M2, M2, 1, M2, 1,
       8 * qfu, qfu, 8 * qfu, qfu, 8 * gu, gu, Bn, H, 1.f, 1.f);
  // complex tanh
  {
    long tot = (long)Bn * H * M * M;
    ctanh_kernel<<<dim3((unsigned)((tot + 255) / 256)), 256, 0, st>>>(G, Bn * H, M);
  }
  // QKV(e,x) = sum_y Kf(e,y) * G(x,y)  (complex; B accessed transposed)
  gemm(st, Kf, G, QKV, nullptr, E, M, M, M2, 1, 1, M2, M2, 1,
       8 * qfu, qfu, 8 * gu, gu, 8 * qfu, qfu, Bn, H, 1.f, 0.f);
  gemm(st, Kf + M, G + M, QKV, nullptr, E, M, M, M2, 1, 1, M2, M2, 1,
       8 * qfu, qfu, 8 * gu, gu, 8 * qfu, qfu, Bn, H, -1.f, 1.f);
  gemm(st, Kf, G + M, QKV + M, nullptr, E, M, M, M2, 1, 1, M2, M2, 1,
       8 * qfu, qfu, 8 * gu, gu, 8 * qfu, qfu, Bn, H, 1.f, 0.f);
  gemm(st, Kf + M, G, QKV + M, nullptr, E, M, M, M2, 1, 1, M2, M2, 1,
       8 * qfu, qfu, 8 * gu, gu, 8 * qfu, qfu, Bn, H, 1.f, 1.f);
  // irfft (+1/4096 scale folded into Fi) with transposed scatter into
  // (b, t, e, h) layout: C(e,t) strides = (8, 512).
  gemm(st, QKV, Fi, outp, nullptr, E, L, M2i, M2, 1, L, 1, 8, 512,
       8 * qfu, qfu, 0, 0, xb, 1, Bn, H, 1.f, beta_out);
}

static inline void fill0(hipStream_t st, float* p, long n)
{
  fill_f32<<<dim3((unsigned)((n + 255) / 256)), 256, 0, st>>>(p, n, 0.f);
}

// ---------------------------------------------------------------------------
extern "C" void kernel_launch(void* const* d_in, const int* in_sizes, int n_in,
                              void* d_out, int out_size, void* d_ws, size_t ws_size,
                              hipStream_t stream)
{
  (void)in_sizes; (void)n_in; (void)out_size; (void)ws_size;
  const float* q    = (const float*)d_in[0];
  const float* k    = (const float*)d_in[1];
  // d_in[2] (v), d_in[7] (Wv), d_in[8] (bv): dead in the reference graph.
  const float* Wq   = (const float*)d_in[3];
  const float* bq   = (const float*)d_in[4];
  const float* Wk   = (const float*)d_in[5];
  const float* bk   = (const float*)d_in[6];
  const float* Wout = (const float*)d_in[9];
  const float* bout = (const float*)d_in[10];
  const float* ec_s = (const float*)d_in[11];
  const float* ec_d = (const float*)d_in[12];
  const float* rc_e = (const float*)d_in[13];
  const float* rc_o = (const float*)d_in[14];
  float* out = (float*)d_out;

  const int Bn = 32, Nn = 1024, Sn = 512, CK = 512;
  long Lsz[10], cum[10];
  {
    long c = 0;
    for (int i = 0; i < 10; ++i) { Lsz[i] = (long)Nn >> (i + 1); cum[i] = c; c += Lsz[i]; }
  }

  // ---- workspace arena (floats) ----
  float* w = (float*)d_ws;
  size_t o = 0;
  auto take = [&](size_t n) { float* p = w + o; o += n; return p; };
  const size_t full = (size_t)Bn * Nn * CK;   // 16.78M floats
  const size_t lvls = (size_t)Bn * 1023 * CK; // all decomposition levels
  float* Qb = take(full);
  float* Kb = take(full);
  float* Dq = take(lvls); float* Sq = take(lvls);
  float* Dk = take(lvls); float* Sk = take(lvls);
  float* Ud = take(lvls); float* Us = take(lvls);
  float* V0 = take(full); float* V1 = take(full);
  const size_t spu = (size_t)Bn * 8 * 64 * 128; // max (b,h,E,2M) plane
  float* Qf = take(spu); float* Kf = take(spu);
  float* Gm = take(spu); float* Qv = take(spu);
  float* Ff[9]; float* Fi[9];
  for (int i = 0; i < 9; ++i) {
    long L = Lsz[i];
    long M = (L / 2 < 64) ? L / 2 : 64;
    Ff[i] = take((size_t)(L * 2 * M));
    Fi[i] = take((size_t)(2 * M * L));
  }

  // ---- twiddle tables (rebuilt every call: deterministic) ----
  for (int i = 0; i < 9; ++i) {
    long L = Lsz[i];
    int  M = (int)((L / 2 < 64) ? L / 2 : 64);
    long tot = L * M;
    build_twiddle<<<dim3((unsigned)((tot + 255) / 256)), 256, 0, stream>>>(
        Ff[i], Fi[i], (int)L, M);
  }

  // ---- projections (k is zero-padded S->N per batch; zero Kb first) ----
  fill0(stream, Kb, (long)full);
  gemm(stream, q, Wq, Qb, bq, Bn * Nn, CK, CK, 512, 1, 512, 1, 512, 1,
       0, 0, 0, 0, 0, 0, 1, 1, 1.f, 0.f);
  gemm(stream, k, Wk, Kb, bk, Sn, CK, CK, 512, 1, 512, 1, 512, 1,
       (long)Sn * CK, 0, 0, 0, (long)Nn * CK, 0, Bn, 1, 1.f, 0.f);

  // ---- wavelet decomposition chains for q and k ----
  const float* cur = Qb;
  for (int i = 0; i < 10; ++i) {
    float* dp = Dq + (size_t)Bn * CK * cum[i];
    float* sp = Sq + (size_t)Bn * CK * cum[i];
    long tot = (long)Bn * Lsz[i] * 64;
    wavelet_step<<<dim3((unsigned)((tot + 255) / 256)), 256, 0, stream>>>(
        cur, dp, sp, ec_d, ec_s, Bn, (int)Lsz[i]);
    cur = sp;
  }
  cur = Kb;
  for (int i = 0; i < 10; ++i) {
    float* dp = Dk + (size_t)Bn * CK * cum[i];
    float* sp = Sk + (size_t)Bn * CK * cum[i];
    long tot = (long)Bn * Lsz[i] * 64;
    wavelet_step<<<dim3((unsigned)((tot + 255) / 256)), 256, 0, stream>>>(
        cur, dp, sp, ec_d, ec_s, Bn, (int)Lsz[i]);
    cur = sp;
  }

  // ---- per-level spectral cross: Us = spec(d); Ud = spec(d) + spec(s) ----
  for (int i = 0; i < 10; ++i) {
    float* usb = Us + (size_t)Bn * CK * cum[i];
    float* udb = Ud + (size_t)Bn * CK * cum[i];
    long   L   = Lsz[i];
    long   n   = (long)Bn * L * CK;
    if (L < 2) { // M = 0: exact zeros
      fill0(stream, usb, n);
      fill0(stream, udb, n);
      continue;
    }
    int M = (int)((L / 2 < 64) ? L / 2 : 64);
    const float* dqb = Dq + (size_t)Bn * CK * cum[i];
    const float* dkb = Dk + (size_t)Bn * CK * cum[i];
    const float* sqb = Sq + (size_t)Bn * CK * cum[i];
    const float* skb = Sk + (size_t)Bn * CK * cum[i];
    spectral_pass(stream, dqb, dkb, usb, 0.f, Qf, Kf, Gm, Qv, Ff[i], Fi[i], (int)L, M);
    copy_f32<<<dim3((unsigned)((n + 255) / 256)), 256, 0, stream>>>(udb, usb, n);
    spectral_pass(stream, sqb, skb, udb, 1.f, Qf, Kf, Gm, Qv, Ff[i], Fi[i], (int)L, M);
  }

  // ---- reconstruction (seed v = spectral at L=1 -> zeros) ----
  fill0(stream, V0, (long)Bn * CK);
  float* vin = V0;
  float* vout = V1;
  for (int i = 9; i >= 0; --i) {
    long L   = Lsz[i];
    long tot = (long)Bn * L * 64;
    recon_step<<<dim3((unsigned)((tot + 255) / 256)), 256, 0, stream>>>(
        vin, Us + (size_t)Bn * CK * cum[i], Ud + (size_t)Bn * CK * cum[i],
        vout, rc_e, rc_o, Bn, (int)L);
    float* tswap = vin; vin = vout; vout = tswap;
  }

  // ---- output projection ----
  gemm(stream, vin, Wout, out, bout, Bn * Nn, CK, CK, 512, 1, 512, 1, 512, 1,
       0, 0, 0, 0, 0, 0, 1, 1, 1.f, 0.f);
}